// NGCF_2319282340320
// MI455X (gfx1250) — compile-verified
//
#include <hip/hip_runtime.h>
#include <hip/hip_bf16.h>

#define N_USER   200000
#define N_ITEM   100000
#define N_NODES  (N_USER + N_ITEM)
#define N_PAD    300032              // padded to multiple of 128 (block M-tile)
#define EMB      80
#define FEAT     16
#define N_LAYERS 3
#define NNZ      1000000
#define BATCH    4096
#define NEG_SLOPE 0.2f
#define EPS_NRM   1e-12f

typedef __attribute__((ext_vector_type(2))) float v2f;
typedef __attribute__((ext_vector_type(8))) float v8f;

// ---------------------------------------------------------------------------
// 1) E[0:N_USER] = user_tab, E[N_USER:N_NODES] = item_tab, pad rows = 0
// ---------------------------------------------------------------------------
__global__ void ngcf_build_E(const float* __restrict__ user_tab,
                             const float* __restrict__ item_tab,
                             float* __restrict__ E) {
    int tid = blockIdx.x * 256 + threadIdx.x;
    const int total = N_PAD * EMB;
    if (tid < total) {
        const int up = N_USER * EMB;
        const int np = N_NODES * EMB;
        float x = 0.0f;
        if (tid < up)      x = user_tab[tid];
        else if (tid < np) x = item_tab[tid - up];
        E[tid] = x;
    }
}

// ---------------------------------------------------------------------------
// 2) Scatter blended batch-user rows into E (last occurrence of u_id wins,
//    deterministic, matching jnp .at[].set semantics).
// ---------------------------------------------------------------------------
__global__ void ngcf_blend_users(const int* __restrict__ u_id,
                                 const int* __restrict__ age,
                                 const int* __restrict__ sex,
                                 const int* __restrict__ month,
                                 const int* __restrict__ day,
                                 const int* __restrict__ dow,
                                 const float* __restrict__ age_tab,
                                 const float* __restrict__ sex_tab,
                                 const float* __restrict__ month_tab,
                                 const float* __restrict__ day_tab,
                                 const float* __restrict__ dow_tab,
                                 const float* __restrict__ user_tab,
                                 float* __restrict__ E) {
    __shared__ int skip;
    const int b = blockIdx.x;
    const int uid = u_id[b];
    if (threadIdx.x == 0) {
        int sk = 0;
        for (int b2 = b + 1; b2 < BATCH; ++b2)
            if (u_id[b2] == uid) { sk = 1; break; }
        skip = sk;
    }
    __syncthreads();
    if (skip) return;
    const int j = threadIdx.x;
    if (j >= EMB) return;
    const int g = j / FEAT, off = j % FEAT;
    float f;
    switch (g) {
        case 0:  f = age_tab[age[b] * FEAT + off];     break;
        case 1:  f = sex_tab[sex[b] * FEAT + off];     break;
        case 2:  f = month_tab[month[b] * FEAT + off]; break;
        case 3:  f = day_tab[day[b] * FEAT + off];     break;
        default: f = dow_tab[dow[b] * FEAT + off];     break;
    }
    E[(size_t)uid * EMB + j] = 0.5f * user_tab[(size_t)uid * EMB + j] + 0.5f * f;
}

// ---------------------------------------------------------------------------
// 3) zero-fill
// ---------------------------------------------------------------------------
__global__ void ngcf_zero(float* __restrict__ p, int n) {
    int tid = blockIdx.x * 256 + threadIdx.x;
    if (tid < n) p[tid] = 0.0f;
}

// ---------------------------------------------------------------------------
// 4) SpMM scatter: L_E[row][j..j+3] += val * E[col][j..j+3]
//    20 threads per nnz, b128 loads of E, 4 f32 atomics (L2-resident).
// ---------------------------------------------------------------------------
__global__ void ngcf_spmm(const int* __restrict__ rows,
                          const int* __restrict__ cols,
                          const float* __restrict__ vals,
                          const float* __restrict__ E,
                          float* __restrict__ LE) {
    int tid = blockIdx.x * 256 + threadIdx.x;
    if (tid >= NNZ * (EMB / 4)) return;
    int e = tid / (EMB / 4);
    int j = (tid - e * (EMB / 4)) * 4;
    int r = rows[e], c = cols[e];
    float v = vals[e];
    const float4 ev = *(const float4*)(E + (size_t)c * EMB + j);
    float* dst = LE + (size_t)r * EMB + j;
    atomicAdd(dst + 0, v * ev.x);
    atomicAdd(dst + 1, v * ev.y);
    atomicAdd(dst + 2, v * ev.z);
    atomicAdd(dst + 3, v * ev.w);
}

// ---------------------------------------------------------------------------
// 5) Layer transform (fp32 WMMA):
//    msg = (LE+E)@W1 + (LE*E)@W2 + (2*b1+b2);  E = leaky_relu(msg) in place.
//    256 threads = 8 waves; each wave owns one 16-row tile and 5 v8f
//    accumulators (full 16x80 output) via v_wmma_f32_16x16x4_f32.
//    A frag (16x4 f32): lane L -> A[m=L%16][kb + (L>=16?2:0) + {0,1}]
//    B frag (4x16 f32): lane L -> B[kb + (L>=16?2:0) + {0,1}][n=L%16]
//    Epilogue stages the activated tile in LDS, then emits coalesced
//    global_store_b128 over the contiguous 1280-float row range.
// ---------------------------------------------------------------------------
__global__ void __launch_bounds__(256)
ngcf_layer_wmma(float* __restrict__ E,
                const float* __restrict__ LE,
                const float* __restrict__ W1,
                const float* __restrict__ b1,
                const float* __restrict__ W2,
                const float* __restrict__ b2) {
    __shared__ float sW1[EMB * EMB];
    __shared__ float sW2[EMB * EMB];
    __shared__ float sBias[EMB];
    __shared__ float sOut[8][16 * EMB];   // per-wave staging, 40 KB

    const int t = threadIdx.x;
    for (int i = t; i < EMB * EMB; i += 256) {
        sW1[i] = W1[i];
        sW2[i] = W2[i];
    }
    if (t < EMB) sBias[t] = 2.0f * b1[t] + b2[t];
    __syncthreads();

    const int wave    = t >> 5;           // 0..7 (wave32)
    const int lane    = t & 31;
    const int rowBase = blockIdx.x * 128 + wave * 16;   // always < N_PAD
    const int mlocal  = lane & 15;
    const int khalf   = (lane >= 16) ? 2 : 0;

    const float* eRow = E  + (size_t)(rowBase + mlocal) * EMB;
    const float* lRow = LE + (size_t)(rowBase + mlocal) * EMB;

    v8f acc[5];
    const v8f vzero = {0.f, 0.f, 0.f, 0.f, 0.f, 0.f, 0.f, 0.f};
#pragma unroll
    for (int nt = 0; nt < 5; ++nt) acc[nt] = vzero;

    for (int kb = 0; kb < EMB; kb += 4) {
        const int k0 = kb + khalf;
        const float e0 = eRow[k0], e1 = eRow[k0 + 1];
        const float l0 = lRow[k0], l1 = lRow[k0 + 1];
        v2f aS = {l0 + e0, l1 + e1};      // (LE + E) fragment
        v2f aP = {l0 * e0, l1 * e1};      // (LE * E) fragment
#pragma unroll
        for (int nt = 0; nt < 5; ++nt) {
            const int ncol = nt * 16 + mlocal;
            v2f bS = {sW1[k0 * EMB + ncol], sW1[(k0 + 1) * EMB + ncol]};
            v2f bP = {sW2[k0 * EMB + ncol], sW2[(k0 + 1) * EMB + ncol]};
            acc[nt] = __builtin_amdgcn_wmma_f32_16x16x4_f32(
                false, aS, false, bS, (short)0, acc[nt], false, false);
            acc[nt] = __builtin_amdgcn_wmma_f32_16x16x4_f32(
                false, aP, false, bP, (short)0, acc[nt], false, false);
        }
    }

    // bias + leaky-ReLU into per-wave LDS staging tile
    const int mhi = (lane >= 16) ? 8 : 0;
    float* stage = &sOut[wave][0];
#pragma unroll
    for (int nt = 0; nt < 5; ++nt) {
        const int ncol = nt * 16 + mlocal;
        const float bias = sBias[ncol];
#pragma unroll
        for (int v = 0; v < 8; ++v) {
            float x = acc[nt][v] + bias;
            x = (x >= 0.0f) ? x : NEG_SLOPE * x;
            stage[(v + mhi) * EMB + ncol] = x;
        }
    }
    __syncthreads();

    // coalesced in-place write-out: 16x80 tile = contiguous 1280 floats
    float4*       dst = (float4*)(E + (size_t)rowBase * EMB);
    const float4* src = (const float4*)stage;
#pragma unroll
    for (int c = 0; c < 10; ++c)
        dst[c * 32 + lane] = src[c * 32 + lane];
}

// ---------------------------------------------------------------------------
// 6) Gather output rows for (u_id, pos_item, neg_item) into slot `slot`
//    of the 320-wide output; optionally L2-normalize the 80-wide row
//    (normalization only matters for the <=12288 gathered rows).
// ---------------------------------------------------------------------------
__global__ void ngcf_gather(const float* __restrict__ E,
                            const int* __restrict__ u_id,
                            const int* __restrict__ pos_item,
                            const int* __restrict__ neg_item,
                            float* __restrict__ out,
                            int slot, int normalize) {
    __shared__ float vals[EMB];
    __shared__ float nrmSh;
    const int blk = blockIdx.x;            // 0 .. 3*BATCH-1
    const int s = blk >> 12;               // /4096
    const int b = blk & 4095;
    int row;
    if (s == 0)      row = u_id[b];
    else if (s == 1) row = N_USER + pos_item[b];
    else             row = N_USER + neg_item[b];

    const int j = threadIdx.x;
    float x = 0.0f;
    if (j < EMB) {
        x = E[(size_t)row * EMB + j];
        vals[j] = x * x;
    }
    __syncthreads();
    if (normalize) {
        if (j == 0) {
            float s2 = 0.0f;
            for (int k = 0; k < EMB; ++k) s2 += vals[k];
            nrmSh = sqrtf(s2);
        }
        __syncthreads();
        x = x / fmaxf(nrmSh, EPS_NRM);
    }
    if (j < EMB)
        out[(size_t)blk * (4 * EMB) + slot * EMB + j] = x;
}

// ---------------------------------------------------------------------------
extern "C" void kernel_launch(void* const* d_in, const int* in_sizes, int n_in,
                              void* d_out, int out_size, void* d_ws, size_t ws_size,
                              hipStream_t stream) {
    const int*   u_id      = (const int*)  d_in[0];
    const int*   age       = (const int*)  d_in[1];
    const int*   sex       = (const int*)  d_in[2];
    const int*   month     = (const int*)  d_in[3];
    const int*   day       = (const int*)  d_in[4];
    const int*   dow       = (const int*)  d_in[5];
    const int*   pos_item  = (const int*)  d_in[6];
    const int*   neg_item  = (const int*)  d_in[7];
    const int*   lap_rows  = (const int*)  d_in[8];
    const int*   lap_cols  = (const int*)  d_in[9];
    const float* lap_vals  = (const float*)d_in[10];
    const float* user_tab  = (const float*)d_in[11];
    const float* item_tab  = (const float*)d_in[12];
    const float* age_tab   = (const float*)d_in[13];
    const float* sex_tab   = (const float*)d_in[14];
    const float* month_tab = (const float*)d_in[15];
    const float* day_tab   = (const float*)d_in[16];
    const float* dow_tab   = (const float*)d_in[17];
    const float* W1        = (const float*)d_in[18];
    const float* b1        = (const float*)d_in[19];
    const float* W2        = (const float*)d_in[20];
    const float* b2        = (const float*)d_in[21];
    float* out = (float*)d_out;

    // Workspace: E (N_PAD rows) | L_E (N_PAD rows). ~192 MB total -> L2.
    float* E  = (float*)d_ws;
    float* LE = E + (size_t)N_PAD * EMB;

    const int nE = N_PAD * EMB;                  // 24,002,560
    const int nScatter = NNZ * (EMB / 4);        // 20,000,000

    ngcf_build_E<<<(nE + 255) / 256, 256, 0, stream>>>(user_tab, item_tab, E);
    ngcf_blend_users<<<BATCH, 128, 0, stream>>>(u_id, age, sex, month, day, dow,
                                                age_tab, sex_tab, month_tab,
                                                day_tab, dow_tab, user_tab, E);
    // slot 0 = un-normalized initial E rows
    ngcf_gather<<<3 * BATCH, 128, 0, stream>>>(E, u_id, pos_item, neg_item, out, 0, 0);

    for (int l = 0; l < N_LAYERS; ++l) {
        ngcf_zero<<<(nE + 255) / 256, 256, 0, stream>>>(LE, nE);
        ngcf_spmm<<<(nScatter + 255) / 256, 256, 0, stream>>>(
            lap_rows, lap_cols, lap_vals, E, LE);
        ngcf_layer_wmma<<<N_PAD / 128, 256, 0, stream>>>(
            E, LE, W1 + (size_t)l * EMB * EMB, b1 + (size_t)l * EMB,
                   W2 + (size_t)l * EMB * EMB, b2 + (size_t)l * EMB);
        ngcf_gather<<<3 * BATCH, 128, 0, stream>>>(E, u_id, pos_item, neg_item,
                                                   out, l + 1, 1);
    }
}